// ProjectedCGCraig_40561671144078
// MI455X (gfx1250) — compile-verified
//
#include <hip/hip_runtime.h>
#include <math.h>

// ---------------- configuration ----------------
#define NBLK  64            // persistent grid: 64 blocks -> safely co-resident
#define TPB   256           // 8 wave32s per block
#define GSZ   (NBLK * TPB)  // 16384 threads
#define NWAVE (GSZ / 32)    // 512 waves

#define M_DIM 1024
#define N_DIM 4096

typedef float v2f __attribute__((ext_vector_type(2)));
typedef float v8f __attribute__((ext_vector_type(8)));

// ---------------- workspace layout (float offsets) ----------------
#define OFF_U     0                 // 1024
#define OFF_TM    1024              // 1024  (uhat scratch)
#define OFF_RHSM  2048              // 1024  (A@w rhs for inner CRAIG)
#define OFF_V     4096              // 4096
#define OFF_TN    8192              // 4096  (Atu / vhat scratch)
#define OFF_XC    12288             // 4096  (inner CRAIG result)
#define OFF_DP    16384             // 4096  (particular solution)
#define OFF_W     20480             // 4096  (w = -g_eff)
#define OFF_R     24576             // 4096
#define OFF_P     28672             // 4096
#define OFF_D     32768             // 4096  (CG iterate)
#define OFF_BPV   36864             // 4096  (H@p)
#define OFF_PBP   40960             // 4096  (projected H@p)
#define OFF_PART  45056             // 8 * 4096 matvec partials
#define OFF_BP    77824             // 8 reduction slots * NBLK block partials
#define OFF_BAR   78336             // 2 ints (barrier count, generation)

// ---------------- grid barrier (device-scope, generation based) ----------------
__device__ __forceinline__ void gbar(int* bar) {
  __syncthreads();
  if (threadIdx.x == 0) {
    int g = __hip_atomic_load(&bar[1], __ATOMIC_ACQUIRE, __HIP_MEMORY_SCOPE_AGENT);
    int prev = __hip_atomic_fetch_add(&bar[0], 1, __ATOMIC_ACQ_REL, __HIP_MEMORY_SCOPE_AGENT);
    if (prev == NBLK - 1) {
      __hip_atomic_store(&bar[0], 0, __ATOMIC_RELAXED, __HIP_MEMORY_SCOPE_AGENT);
      __hip_atomic_fetch_add(&bar[1], 1, __ATOMIC_ACQ_REL, __HIP_MEMORY_SCOPE_AGENT);
    } else {
      while (__hip_atomic_load(&bar[1], __ATOMIC_ACQUIRE, __HIP_MEMORY_SCOPE_AGENT) == g) {
        __builtin_amdgcn_s_sleep(1);
      }
    }
  }
  __syncthreads();
}

// ---------------- deterministic block reduction -> per-block partials ----------------
__device__ __forceinline__ void block_reduce2(float a, float b2, float* bp0, float* bp1) {
  __shared__ float s0[8], s1[8];
  #pragma unroll
  for (int o = 16; o > 0; o >>= 1) {
    a  += __shfl_down(a,  o, 32);
    b2 += __shfl_down(b2, o, 32);
  }
  int wv = threadIdx.x >> 5;
  __syncthreads();                       // protect s0/s1 from previous call
  if ((threadIdx.x & 31) == 0) { s0[wv] = a; s1[wv] = b2; }
  __syncthreads();
  if (threadIdx.x == 0) {
    float t0 = 0.f, t1 = 0.f;
    #pragma unroll
    for (int i = 0; i < 8; ++i) { t0 += s0[i]; t1 += s1[i]; }
    bp0[blockIdx.x] = t0;
    if (bp1) bp1[blockIdx.x] = t1;
  }
}

// every thread redundantly (and identically) sums the block partials
__device__ __forceinline__ float sum_bp(const float* __restrict__ q) {
  float t = 0.f;
  #pragma unroll
  for (int i = 0; i < NBLK; ++i) t += q[i];
  return t;
}

// ---------------- WMMA fp32 matvec: y = M(RxC,row-major) @ x ----------------
// A-operand: 16x4 tile of M (ISA layout: lanes0-15 K=0,1 ; lanes16-31 K=2,3).
// B-operand: x segment in column N=0 only (lanes 0 and 16), built branchlessly
// with a cndmask select so no exec-mask branching appears in the inner loop.
// Wave-uniform base + 32-bit per-lane offset -> global_load saddr form.
__device__ __forceinline__ void mv_rows_wmma(const float* __restrict__ Mm,
                                             const float* __restrict__ x,
                                             float* __restrict__ part, int R, int C, int nsl,
                                             int wgid, int lane) {
  const int  ntiles = R >> 4;
  const int  jobs   = ntiles * nsl;
  const int  klen   = C / nsl;
  const int  mrow   = lane & 15;
  const int  kofs   = (lane >> 4) << 1;     // 0 or 2
  const bool col0   = (mrow == 0);          // lanes 0 and 16 carry B column 0
  const v2f  zz     = {0.f, 0.f};
  for (int job = wgid; job < jobs; job += NWAVE) {
    const int tile = job % ntiles, sl = job / ntiles;
    const int r0 = tile << 4;
    const int kb = sl * klen, ke = kb + klen;
    const float* tilep = Mm + (size_t)r0 * (size_t)C;  // wave-uniform base
    int aidx = mrow * C + kb + kofs;                   // per-lane 32-bit offset
    v8f c = {0.f, 0.f, 0.f, 0.f, 0.f, 0.f, 0.f, 0.f};
    #pragma unroll 4
    for (int k = kb; k < ke; k += 4) {
      v2f a  = *(const v2f*)(tilep + aidx);            // A[M=mrow][K=k+kofs..]
      v2f xv = *(const v2f*)(x + k + kofs);            // coalesced, in-bounds
      v2f bf = col0 ? xv : zz;                         // B[.][0] only
      c = __builtin_amdgcn_wmma_f32_16x16x4_f32(false, a, false, bf, (short)0, c,
                                                false, false);
      aidx += 4;
    }
    // D[m][0]: lane0 holds rows r0..r0+7, lane16 rows r0+8..r0+15
    if (col0) {
      float* dst = part + (size_t)sl * R + r0 + ((lane >> 4) << 3);
      #pragma unroll
      for (int rr = 0; rr < 8; ++rr) dst[rr] = c[rr];
    }
  }
}

// ---------------- WMMA fp32 transposed matvec: y = A^T u (A: M_DIM x N_DIM) ----------------
// u replicated in row M=0 of the A-operand (lanes 0/16, branchless select);
// 4x16 tile of A in the B-operand via coalesced row-major loads.
__device__ __forceinline__ void mvT_wmma(const float* __restrict__ Am,
                                         const float* __restrict__ x,
                                         float* __restrict__ part, int nsl,
                                         int wgid, int lane) {
  const int  ntiles = N_DIM >> 4;
  const int  jobs   = ntiles * nsl;
  const int  ilen   = M_DIM / nsl;
  const int  nlane  = lane & 15;
  const int  krow   = (lane >> 4) << 1;     // B rows 0,1 (lanes<16) or 2,3
  const bool lead   = (nlane == 0);         // lanes 0 and 16 carry A row 0
  const v2f  zz     = {0.f, 0.f};
  for (int job = wgid; job < jobs; job += NWAVE) {
    const int tile = job % ntiles, sl = job / ntiles;
    const int j0 = tile << 4;
    const int ib = sl * ilen, ie = ib + ilen;
    int bidx = (ib + krow) * N_DIM + j0 + nlane;       // per-lane 32-bit offset
    v8f c = {0.f, 0.f, 0.f, 0.f, 0.f, 0.f, 0.f, 0.f};
    #pragma unroll 4
    for (int i = ib; i < ie; i += 4) {
      v2f bf;
      bf.x = Am[bidx];                                  // B[krow  ][n]
      bf.y = Am[bidx + N_DIM];                          // B[krow+1][n]
      v2f xv = *(const v2f*)(x + i + krow);             // u segment
      v2f a  = lead ? xv : zz;                          // A row 0 only
      c = __builtin_amdgcn_wmma_f32_16x16x4_f32(false, a, false, bf, (short)0, c,
                                                false, false);
      bidx += 4 * N_DIM;
    }
    if (lane < 16) part[(size_t)sl * N_DIM + j0 + lane] = c[0];   // D[0][n]
  }
}

// ---------------- CRAIG min-norm solve of A x = rhs (faithful to reference) ----------------
__device__ __forceinline__ void craig_solve(const float* __restrict__ A,
                                            const float* __restrict__ rhs,
                                            float* __restrict__ x,
                                            float* __restrict__ u, float* __restrict__ v,
                                            float* __restrict__ tn, float* __restrict__ tm,
                                            float* __restrict__ part, float* __restrict__ bp,
                                            int* bar, int tid, int lane, int wgid) {
  { float acc = 0.f;
    for (int i = tid; i < M_DIM; i += GSZ) acc += rhs[i] * rhs[i];
    block_reduce2(acc, 0.f, bp + 0 * NBLK, nullptr); }
  gbar(bar);
  const float beta1  = sqrtf(sum_bp(bp + 0 * NBLK));
  const float b1safe = fmaxf(beta1, 1e-30f);
  const float thr    = fmaxf(1e-12f, 1e-10f * b1safe);
  const bool  trivially = beta1 < 1e-10f * fmaxf(b1safe, 1.0f);
  for (int i = tid; i < M_DIM; i += GSZ) u[i] = rhs[i] / b1safe;
  gbar(bar);
  mvT_wmma(A, u, part, 2, wgid, lane);
  gbar(bar);
  { float acc = 0.f;
    for (int j = tid; j < N_DIM; j += GSZ) {
      float t = part[j] + part[N_DIM + j];
      tn[j] = t; acc += t * t;
    }
    block_reduce2(acc, 0.f, bp + 1 * NBLK, nullptr); }
  gbar(bar);
  const float alpha1 = sqrtf(sum_bp(bp + 1 * NBLK));
  const bool  bdown  = alpha1 < 1e-14f;
  const float a1safe = fmaxf(alpha1, 1e-30f);
  float s = beta1 / a1safe;
  for (int j = tid; j < N_DIM; j += GSZ) {
    float vv = tn[j] / a1safe;
    v[j] = vv;
    x[j] = bdown ? 0.f : s * vv;
  }
  gbar(bar);
  mv_rows_wmma(A, v, part, M_DIM, N_DIM, 8, wgid, lane);
  gbar(bar);
  { float acc = 0.f;
    for (int i = tid; i < M_DIM; i += GSZ) {
      float t = 0.f;
      #pragma unroll
      for (int sl = 0; sl < 8; ++sl) t += part[sl * M_DIM + i];
      t -= alpha1 * u[i];
      tm[i] = t; acc += t * t;
    }
    block_reduce2(acc, 0.f, bp + 2 * NBLK, nullptr); }
  gbar(bar);
  float beta = sqrtf(sum_bp(bp + 2 * NBLK));
  bool converged = trivially || (fabsf(beta * s) < thr) || (bdown && !trivially);
  { float usc = 1.f / fmaxf(beta, 1e-30f);
    for (int i = tid; i < M_DIM; i += GSZ) u[i] = tm[i] * usc; }
  gbar(bar);

  for (int it = 0; it < 30; ++it) {
    if (converged) break;                          // uniform across all blocks
    mvT_wmma(A, u, part, 2, wgid, lane);
    gbar(bar);
    { float acc = 0.f;
      for (int j = tid; j < N_DIM; j += GSZ) {
        float t = part[j] + part[N_DIM + j] - beta * v[j];
        tn[j] = t; acc += v[j] * t;                // dot(v, vhat)
      }
      block_reduce2(acc, 0.f, bp + 3 * NBLK, nullptr); }
    gbar(bar);
    const float dvv = sum_bp(bp + 3 * NBLK);
    { float acc = 0.f;
      for (int j = tid; j < N_DIM; j += GSZ) {
        float t = tn[j] - dvv * v[j];              // reorthogonalize
        tn[j] = t; acc += t * t;
      }
      block_reduce2(acc, 0.f, bp + 4 * NBLK, nullptr); }
    gbar(bar);
    const float alpha = sqrtf(sum_bp(bp + 4 * NBLK));
    const bool  abrk  = alpha < 1e-14f;
    const float asafe = fmaxf(alpha, 1e-30f);
    const float snew  = -beta * s / asafe;
    for (int j = tid; j < N_DIM; j += GSZ) {
      float vv = tn[j] / asafe;
      v[j] = vv;
      if (!abrk) x[j] += snew * vv;
    }
    s = snew;
    gbar(bar);
    mv_rows_wmma(A, v, part, M_DIM, N_DIM, 8, wgid, lane);
    gbar(bar);
    { float acc = 0.f;
      for (int i = tid; i < M_DIM; i += GSZ) {
        float t = 0.f;
        #pragma unroll
        for (int sl = 0; sl < 8; ++sl) t += part[sl * M_DIM + i];
        t -= alpha * u[i];
        tm[i] = t; acc += u[i] * t;                // dot(u, uhat)
      }
      block_reduce2(acc, 0.f, bp + 5 * NBLK, nullptr); }
    gbar(bar);
    const float duu = sum_bp(bp + 5 * NBLK);
    { float acc = 0.f;
      for (int i = tid; i < M_DIM; i += GSZ) {
        float t = tm[i] - duu * u[i];
        tm[i] = t; acc += t * t;
      }
      block_reduce2(acc, 0.f, bp + 6 * NBLK, nullptr); }
    gbar(bar);
    const float bnew = sqrtf(sum_bp(bp + 6 * NBLK));
    const bool  bbrk = bnew < 1e-14f;
    converged = (fabsf(bnew * s) < thr) || abrk || bbrk;
    beta = bnew;
    { float usc = 1.f / fmaxf(bnew, 1e-30f);
      for (int i = tid; i < M_DIM; i += GSZ) u[i] = tm[i] * usc; }
    gbar(bar);
  }
}

// ---------------- persistent kernel: whole algorithm in one launch ----------------
__global__ void __launch_bounds__(TPB)
pcg_persistent(const float* __restrict__ A, const float* __restrict__ b,
               const float* __restrict__ g, const float* __restrict__ H,
               float* __restrict__ out, float* __restrict__ wsf, int* bar) {
  const int tid  = blockIdx.x * TPB + threadIdx.x;
  const int lane = threadIdx.x & 31;
  const int wgid = tid >> 5;

  float* u    = wsf + OFF_U;
  float* tm   = wsf + OFF_TM;
  float* rhsm = wsf + OFF_RHSM;
  float* v    = wsf + OFF_V;
  float* tn   = wsf + OFF_TN;
  float* xc   = wsf + OFF_XC;
  float* dp   = wsf + OFF_DP;
  float* w    = wsf + OFF_W;
  float* r    = wsf + OFF_R;
  float* p    = wsf + OFF_P;
  float* dv   = wsf + OFF_D;
  float* Bp   = wsf + OFF_BPV;
  float* PBp  = wsf + OFF_PBP;
  float* part = wsf + OFF_PART;
  float* bp   = wsf + OFF_BP;

  // d_p = craig_solve(A, b)
  craig_solve(A, b, dp, u, v, tn, tm, part, bp, bar, tid, lane, wgid);

  // w = -(g + H @ d_p)
  mv_rows_wmma(H, dp, part, N_DIM, N_DIM, 2, wgid, lane);
  gbar(bar);
  for (int j = tid; j < N_DIM; j += GSZ)
    w[j] = -(g[j] + part[j] + part[N_DIM + j]);
  gbar(bar);

  // r0 = w - craig(A, A @ w)
  mv_rows_wmma(A, w, part, M_DIM, N_DIM, 8, wgid, lane);
  gbar(bar);
  for (int i = tid; i < M_DIM; i += GSZ) {
    float t = 0.f;
    #pragma unroll
    for (int sl = 0; sl < 8; ++sl) t += part[sl * M_DIM + i];
    rhsm[i] = t;
  }
  gbar(bar);
  craig_solve(A, rhsm, xc, u, v, tn, tm, part, bp, bar, tid, lane, wgid);
  { float acc = 0.f;
    for (int j = tid; j < N_DIM; j += GSZ) {
      float rr = w[j] - xc[j];
      r[j] = rr; p[j] = rr; dv[j] = 0.f;
      acc += rr * rr;
    }
    block_reduce2(acc, 0.f, bp + 7 * NBLK, nullptr); }
  gbar(bar);
  float rz = sum_bp(bp + 7 * NBLK);
  bool cgdone = rz < 1e-16f;                       // CG_TOL^2

  for (int cg = 0; cg < 20; ++cg) {
    if (cgdone) break;
    // Bp = H @ p
    mv_rows_wmma(H, p, part, N_DIM, N_DIM, 2, wgid, lane);
    gbar(bar);
    for (int j = tid; j < N_DIM; j += GSZ)
      Bp[j] = part[j] + part[N_DIM + j];
    gbar(bar);
    // PBp = Bp - craig(A, A @ Bp)
    mv_rows_wmma(A, Bp, part, M_DIM, N_DIM, 8, wgid, lane);
    gbar(bar);
    for (int i = tid; i < M_DIM; i += GSZ) {
      float t = 0.f;
      #pragma unroll
      for (int sl = 0; sl < 8; ++sl) t += part[sl * M_DIM + i];
      rhsm[i] = t;
    }
    gbar(bar);
    craig_solve(A, rhsm, xc, u, v, tn, tm, part, bp, bar, tid, lane, wgid);
    { float a1 = 0.f, a2 = 0.f;
      for (int j = tid; j < N_DIM; j += GSZ) {
        float t = Bp[j] - xc[j];
        PBp[j] = t;
        a1 += p[j] * t;
        a2 += p[j] * p[j];
      }
      block_reduce2(a1, a2, bp + 0 * NBLK, bp + 1 * NBLK); }
    gbar(bar);
    const float pPBp = sum_bp(bp + 0 * NBLK);
    const float pp   = sum_bp(bp + 1 * NBLK);
    if (pPBp <= 1e-6f * pp) break;                 // curvature guard: revert + stop
    const float alpha = rz / fmaxf(pPBp, 1e-30f);
    { float acc = 0.f;
      for (int j = tid; j < N_DIM; j += GSZ) {
        dv[j] += alpha * p[j];
        float rr = r[j] - alpha * PBp[j];
        r[j] = rr;
        acc += rr * rr;
      }
      block_reduce2(acc, 0.f, bp + 2 * NBLK, nullptr); }
    gbar(bar);
    const float rn2 = sum_bp(bp + 2 * NBLK);
    const float cb  = rn2 / fmaxf(rz, 1e-30f);
    for (int j = tid; j < N_DIM; j += GSZ)
      p[j] = r[j] + cb * p[j];
    gbar(bar);
    rz = rn2;
    cgdone = rn2 < 1e-16f;
  }

  for (int j = tid; j < N_DIM; j += GSZ)
    out[j] = dp[j] + dv[j];
}

__global__ void init_bar_kernel(int* bar) {
  if (threadIdx.x < 2) bar[threadIdx.x] = 0;
}

extern "C" void kernel_launch(void* const* d_in, const int* in_sizes, int n_in,
                              void* d_out, int out_size, void* d_ws, size_t ws_size,
                              hipStream_t stream) {
  (void)in_sizes; (void)n_in; (void)out_size; (void)ws_size;
  const float* A = (const float*)d_in[0];
  const float* b = (const float*)d_in[1];
  const float* g = (const float*)d_in[2];
  const float* H = (const float*)d_in[3];
  float* out = (float*)d_out;
  float* wsf = (float*)d_ws;
  int*   bar = (int*)(wsf + OFF_BAR);

  init_bar_kernel<<<1, 32, 0, stream>>>(bar);
  pcg_persistent<<<NBLK, TPB, 0, stream>>>(A, b, g, H, out, wsf, bar);
}